// SVAE_LDS_31568009625999
// MI455X (gfx1250) — compile-verified
//
#include <hip/hip_runtime.h>
#include <stdint.h>

// ---------------- problem constants ----------------
#define DD     4
#define POSN   2
#define TT     2048
#define KOBS   128
#define TM1    2047
#define ITERS  5
#define LOG2PI_F 1.8378770664093453f
// P_D=0.9, G=0.01, M_T=128 -> P_D/M_T = 0.00703125, (1-P_D)*G = 0.001

// output flat offsets (return order: x_recon, z_recon, mq, Pq, mf, Pf, ll)
#define OUT_XREC 0
#define OUT_ZREC 4094
#define OUT_MQ   12282
#define OUT_PQ   20470
#define OUT_MF   53222
#define OUT_PF   61410
#define OUT_LL   94162

// workspace offsets (floats)
#define WS_XHAT 0u                      // T*K*D = 1048576
#define WS_BETA (WS_XHAT + 1048576u)    // Tm1*(K+1) = 264063
#define WS_MF   (WS_BETA + 264063u)     // 8188
#define WS_PF   (WS_MF   + 8188u)       // 32752
#define WS_MQ   (WS_PF   + 32752u)      // 8188
#define WS_PQ   (WS_MQ   + 8188u)       // 32752
#define WS_CST  (WS_PQ   + 32752u)      // 96
#define WS_DUM  (WS_CST  + 96u)         // 1

// const-block layout
#define CST_A    0
#define CST_B    16
#define CST_Q    20
#define CST_R    36
#define CST_QINV 52
#define CST_CINV 68
#define CST_CLP  84

typedef __attribute__((ext_vector_type(2))) float v2f;
typedef __attribute__((ext_vector_type(8))) float v8f;

// ---------------- small 4x4 helpers ----------------
__device__ inline void mm4(const float* A, const float* B, float* C) {
  for (int i = 0; i < 4; ++i)
    for (int j = 0; j < 4; ++j) {
      float s = 0.f;
      for (int k = 0; k < 4; ++k) s += A[i*4+k] * B[k*4+j];
      C[i*4+j] = s;
    }
}

__device__ inline void chol4(const float* A, float* L) {
  for (int i = 0; i < 16; ++i) L[i] = 0.f;
  for (int j = 0; j < 4; ++j) {
    float s = A[j*4+j];
    for (int k = 0; k < j; ++k) s -= L[j*4+k] * L[j*4+k];
    float d = sqrtf(fmaxf(s, 1e-20f));
    L[j*4+j] = d;
    float inv = 1.f / d;
    for (int i = j + 1; i < 4; ++i) {
      float t = A[i*4+j];
      for (int k = 0; k < j; ++k) t -= L[i*4+k] * L[j*4+k];
      L[i*4+j] = t * inv;
    }
  }
}

// SPD inverse via Cholesky; optional sum(log diag(L))
__device__ inline void spd_inv4(const float* A, float* Ainv, float* sumlogdiag) {
  float L[16];
  chol4(A, L);
  float Li[16];
  for (int i = 0; i < 16; ++i) Li[i] = 0.f;
  for (int i = 0; i < 4; ++i) {
    float di = 1.f / L[i*4+i];
    Li[i*4+i] = di;
    for (int j = 0; j < i; ++j) {
      float s = 0.f;
      for (int k = j; k < i; ++k) s += L[i*4+k] * Li[k*4+j];
      Li[i*4+j] = -s * di;
    }
  }
  for (int i = 0; i < 4; ++i)
    for (int j = 0; j < 4; ++j) {
      int k0 = (i > j) ? i : j;
      float s = 0.f;
      for (int k = k0; k < 4; ++k) s += Li[k*4+i] * Li[k*4+j];
      Ainv[i*4+j] = s;
    }
  if (sumlogdiag) {
    float s = 0.f;
    for (int i = 0; i < 4; ++i) s += logf(L[i*4+i]);
    *sumlogdiag = s;
  }
}

__device__ inline void vec2cov(const float* v, float* Cov) {
  const int rows[10] = {0,1,1,2,2,2,3,3,3,3};
  const int cols[10] = {0,0,1,0,1,2,0,1,2,3};
  float L[16];
  for (int i = 0; i < 16; ++i) L[i] = 0.f;
  for (int k = 0; k < 10; ++k) L[rows[k]*4+cols[k]] = v[k];
  for (int i = 0; i < 4; ++i) L[i*4+i] = expf(L[i*4+i]);
  for (int i = 0; i < 4; ++i)
    for (int j = 0; j < 4; ++j) {
      float s = 0.f;
      for (int k = 0; k < 4; ++k) s += L[i*4+k] * L[j*4+k];
      Cov[i*4+j] = s;
    }
}

// ---------------- wave32 reductions ----------------
__device__ inline float wave_sum(float v) {
  for (int o = 16; o > 0; o >>= 1) v += __shfl_xor(v, o, 32);
  return v;
}
__device__ inline float wave_max(float v) {
  for (int o = 16; o > 0; o >>= 1) v = fmaxf(v, __shfl_xor(v, o, 32));
  return v;
}

// ---------------- threefry2x32 + Box-Muller (eps ~ N(0,1)) ----------------
__device__ inline unsigned rotl32(unsigned x, int r) { return (x << r) | (x >> (32 - r)); }
__device__ inline void threefry2x32(unsigned k0, unsigned k1, unsigned c0, unsigned c1,
                                    unsigned& o0, unsigned& o1) {
  const int R[8] = {13, 15, 26, 6, 17, 29, 16, 24};
  unsigned ks[3] = {k0, k1, 0x1BD11BDAu ^ k0 ^ k1};
  unsigned x0 = c0 + ks[0], x1 = c1 + ks[1];
  for (int g = 0; g < 5; ++g) {
    for (int i = 0; i < 4; ++i) {
      x0 += x1;
      x1 = rotl32(x1, R[(g * 4 + i) & 7]);
      x1 ^= x0;
    }
    x0 += ks[(g + 1) % 3];
    x1 += ks[(g + 2) % 3] + (unsigned)(g + 1);
  }
  o0 = x0; o1 = x1;
}
__device__ inline void boxmuller(unsigned w0, unsigned w1, float& n0, float& n1) {
  float u1 = ((float)((w0 >> 8) + 1u)) * (1.0f / 16777216.0f);   // (0,1]
  float u2 = ((float)(w1 >> 8)) * (1.0f / 16777216.0f) * 6.28318530717958647f;
  float r = sqrtf(-2.0f * logf(u1));
  n0 = r * __cosf(u2);
  n1 = r * __sinf(u2);
}

// ---------------- kernel: build Q,R,Qinv,C,Cinv consts ----------------
__global__ void svae_setup_kernel(const float* __restrict__ A, const float* __restrict__ b,
                                  const float* __restrict__ Qp, const float* __restrict__ Rp,
                                  float* __restrict__ cst) {
  if (threadIdx.x != 0 || blockIdx.x != 0) return;
  float Q[16], Rm[16];
  vec2cov(Qp, Q);
  vec2cov(Rp, Rm);
  float Qinv[16];
  spd_inv4(Q, Qinv, nullptr);
  float C[16];
  for (int i = 0; i < 16; ++i) C[i] = Q[i] + Rm[i];
  float Cinv[16], sld;
  spd_inv4(C, Cinv, &sld);
  for (int i = 0; i < 16; ++i) {
    cst[CST_A + i] = A[i];
    cst[CST_Q + i] = Q[i];
    cst[CST_R + i] = Rm[i];
    cst[CST_QINV + i] = Qinv[i];
    cst[CST_CINV + i] = Cinv[i];
  }
  for (int i = 0; i < 4; ++i) cst[CST_B + i] = b[i];
  cst[CST_CLP] = -sld - 2.0f * LOG2PI_F;   // -sum(log diag Lc) - 0.5*D*log(2pi)
}

// ---------------- kernel: encoder xhat = x @ W + b via WMMA f32 16x16x4 ----------------
// One wave per 16-row tile; A = 16x4 (x padded 2->4), B = 4x16 (W padded), f32 exact.
__global__ __launch_bounds__(256) void svae_enc_wmma_kernel(
    const float* __restrict__ x, const float* __restrict__ W,
    const float* __restrict__ bvec, float* __restrict__ xhat) {
  int tid = threadIdx.x;
  int wv = tid >> 5, lane = tid & 31, n = lane & 15;
  int tile = blockIdx.x * 8 + wv;          // 16384 tiles total
  int row0 = tile * 16;
  // A: VGPR0 = (M=lane, K=0 | K=2), VGPR1 = (K=1 | K=3); K>=2 is zero pad
  v2f a;
  if (lane < 16) {
    const float* xr = x + (size_t)(row0 + n) * POSN;
    a.x = xr[0];
    a.y = xr[1];
  } else {
    a.x = 0.f; a.y = 0.f;
  }
  // B: 4x16, rows 0,1 = W rows (POS=2), rows 2,3 zero; cols>=4 zero
  v2f bm;
  if (lane < 16) {
    bm.x = (n < DD) ? W[n] : 0.f;        // k=0 row
    bm.y = (n < DD) ? W[DD + n] : 0.f;   // k=1 row
  } else {
    bm.x = 0.f; bm.y = 0.f;              // k=2,3 rows (zero pad)
  }
  v8f c;
  float bias = (n < DD) ? bvec[n] : 0.f;
  for (int j = 0; j < 8; ++j) c[j] = bias;
  c = __builtin_amdgcn_wmma_f32_16x16x4_f32(false, a, false, bm, (short)0, c, false, false);
  // D layout: vgpr j -> M=j (lanes 0-15) / M=j+8 (lanes 16-31), N = lane&15
  if (n < DD) {
    int base = row0 + ((lane < 16) ? 0 : 8);
    for (int j = 0; j < 8; ++j) xhat[(size_t)(base + j) * DD + n] = c[j];
  }
}

// ---------------- kernel: init beta0 ----------------
__global__ void svae_init_beta_kernel(float* __restrict__ beta) {
  int i = blockIdx.x * blockDim.x + threadIdx.x;
  if (i >= TM1 * (KOBS + 1)) return;
  beta[i] = (i % (KOBS + 1) == 0) ? 0.1f : (0.9f / 128.0f);
}

// ---------------- kernel: forward Kalman scan (1 block, 128 threads) ----------------
__global__ __launch_bounds__(128) void svae_fwd_kernel(
    const float* __restrict__ xhat, const float* __restrict__ beta,
    const float* __restrict__ cst, float* __restrict__ mf, float* __restrict__ pf,
    float* __restrict__ ll_out) {
  int tid = threadIdx.x;
  int wv = tid >> 5, ln = tid & 31;
  __shared__ float sA[16], sb[4], sQ[16], sR[16];
  __shared__ float s_mu[4], s_P[16];
  __shared__ float s_red[4][12];
  __shared__ float s_res[12];
  __shared__ float s_ll;

  if (tid < 16) {
    sA[tid] = cst[CST_A + tid];
    sQ[tid] = cst[CST_Q + tid];
    sR[tid] = cst[CST_R + tid];
    s_P[tid] = cst[CST_R + tid];      // P0 = R
  }
  if (tid < 4) {
    sb[tid] = cst[CST_B + tid];
    s_mu[tid] = xhat[tid];            // mu0 = xhat[0,0,:]
  }
  if (tid == 0) s_ll = 0.f;
  __syncthreads();

  for (int t = 0; t < TM1; ++t) {
    // snapshot state + consts into registers (redundant per lane: tiny 4x4 algebra)
    float mu[4], P[16], Ar[16], Qr[16], Rr[16], br[4];
    for (int i = 0; i < 16; ++i) { Ar[i] = sA[i]; Qr[i] = sQ[i]; Rr[i] = sR[i]; P[i] = s_P[i]; }
    for (int i = 0; i < 4; ++i)  { br[i] = sb[i]; mu[i] = s_mu[i]; }

    // predict
    float mup[4];
    for (int i = 0; i < 4; ++i) {
      float s = br[i];
      for (int j = 0; j < 4; ++j) s += Ar[i*4+j] * mu[j];
      mup[i] = s;
    }
    float AP[16], Pp[16];
    mm4(Ar, P, AP);
    for (int i = 0; i < 4; ++i)
      for (int j = 0; j < 4; ++j) {
        float s = Qr[i*4+j];
        for (int k = 0; k < 4; ++k) s += AP[i*4+k] * Ar[j*4+k];
        Pp[i*4+j] = s;
      }
    float S[16];
    for (int i = 0; i < 16; ++i) S[i] = Pp[i] + Rr[i];
    float Sinv[16], sld;
    spd_inv4(S, Sinv, &sld);
    float Kg[16];
    mm4(Pp, Sinv, Kg);
    float Pk[16];
    for (int i = 0; i < 4; ++i)
      for (int j = 0; j < 4; ++j) {
        float s = 0.f;
        for (int k = 0; k < 4; ++k) s += Kg[i*4+k] * Pp[k*4+j];
        Pk[i*4+j] = Pp[i*4+j] - s;
      }

    // per-observation work (lane = k)
    const float* yk = xhat + (size_t)(t + 1) * KOBS * DD + tid * DD;
    float inn[4];
    for (int j = 0; j < 4; ++j) inn[j] = yk[j] - mup[j];
    if (t + 2 < TT)  // warm next step's observations (global_prefetch_b8)
      __builtin_prefetch(xhat + (size_t)(t + 2) * KOBS * DD + tid * DD, 0, 1);
    float quad = 0.f;
    for (int i = 0; i < 4; ++i) {
      float s = 0.f;
      for (int j = 0; j < 4; ++j) s += Sinv[i*4+j] * inn[j];
      quad += inn[i] * s;
    }
    float llk = -0.5f * quad - sld - 2.0f * LOG2PI_F;
    float muk[4];
    for (int i = 0; i < 4; ++i) {
      float s = mup[i];
      for (int j = 0; j < 4; ++j) s += Kg[i*4+j] * inn[j];
      muk[i] = s;
    }
    float b0 = beta[(size_t)t * (KOBS + 1)];
    float bk = beta[(size_t)t * (KOBS + 1) + 1 + tid];
    float term = logf(bk + 1e-30f) + llk;
    float lb0g = logf(b0 * 0.01f);

    // phase A: sum {bk, bk*muk[0..3]}, max(term)
    float v0 = wave_sum(bk);
    float v1 = wave_sum(bk * muk[0]);
    float v2 = wave_sum(bk * muk[1]);
    float v3 = wave_sum(bk * muk[2]);
    float v4 = wave_sum(bk * muk[3]);
    float v5 = wave_max(term);
    if (ln == 0) {
      s_red[wv][0] = v0; s_red[wv][1] = v1; s_red[wv][2] = v2;
      s_red[wv][3] = v3; s_red[wv][4] = v4; s_red[wv][5] = v5;
    }
    __syncthreads();
    if (tid < 5) s_res[tid] = s_red[0][tid] + s_red[1][tid] + s_red[2][tid] + s_red[3][tid];
    if (tid == 5) {
      float m = fmaxf(fmaxf(s_red[0][5], s_red[1][5]), fmaxf(s_red[2][5], s_red[3][5]));
      s_res[5] = fmaxf(m, lb0g);
    }
    __syncthreads();
    float sumbk = s_res[0];
    float muf[4];
    for (int i = 0; i < 4; ++i) muf[i] = b0 * mup[i] + s_res[1 + i];
    float M = s_res[5];
    float e = expf(term - M);
    float dk[4];
    for (int i = 0; i < 4; ++i) dk[i] = muk[i] - muf[i];

    // phase B: 10 weighted-outer sums + sum(exp)
    float w[11];
    int c = 0;
    for (int i = 0; i < 4; ++i)
      for (int j = i; j < 4; ++j) w[c++] = bk * dk[i] * dk[j];
    w[10] = e;
    for (int q = 0; q < 11; ++q) {
      float vv = wave_sum(w[q]);
      if (ln == 0) s_red[wv][q] = vv;
    }
    __syncthreads();
    if (tid < 11) s_res[tid] = s_red[0][tid] + s_red[1][tid] + s_red[2][tid] + s_red[3][tid];
    __syncthreads();

    float d0[4];
    for (int i = 0; i < 4; ++i) d0[i] = mup[i] - muf[i];
    float xx[16];
    int c2 = 0;
    for (int i = 0; i < 4; ++i)
      for (int j = i; j < 4; ++j) { xx[i*4+j] = s_res[c2]; xx[j*4+i] = s_res[c2]; c2++; }
    float Pf_[16];
    for (int i = 0; i < 4; ++i)
      for (int j = 0; j < 4; ++j)
        Pf_[i*4+j] = b0 * (Pp[i*4+j] + d0[i] * d0[j]) + sumbk * Pk[i*4+j] + xx[i*4+j];
    if (tid == 0) {
      float esum = s_res[10];
      s_ll += M + logf(esum + expf(lb0g - M));
    }
    if (tid < 4)  { mf[(size_t)t * 4 + tid]  = muf[tid]; s_mu[tid] = muf[tid]; }
    if (tid < 16) { pf[(size_t)t * 16 + tid] = Pf_[tid]; s_P[tid]  = Pf_[tid]; }
    __syncthreads();
  }
  if (tid == 0) ll_out[0] = s_ll;
}

// ---------------- kernel: backward RTS smoother (serial 4x4 chain) ----------------
__global__ void svae_bwd_kernel(const float* __restrict__ mf, const float* __restrict__ pf,
                                const float* __restrict__ cst,
                                float* __restrict__ mq, float* __restrict__ pq) {
  if (threadIdx.x != 0 || blockIdx.x != 0) return;
  float A[16], b[4], Q[16];
  for (int i = 0; i < 16; ++i) { A[i] = cst[CST_A + i]; Q[i] = cst[CST_Q + i]; }
  for (int i = 0; i < 4; ++i) b[i] = cst[CST_B + i];

  int last = TM1 - 1;
  float mus[4], Ps[16];
  for (int i = 0; i < 4; ++i)  { mus[i] = mf[(size_t)last * 4 + i]; mq[(size_t)last * 4 + i] = mus[i]; }
  for (int i = 0; i < 16; ++i) { Ps[i] = pf[(size_t)last * 16 + i]; pq[(size_t)last * 16 + i] = Ps[i]; }

  for (int t = last - 1; t >= 0; --t) {
    float muf[4], Pf[16];
    for (int i = 0; i < 4; ++i)  muf[i] = mf[(size_t)t * 4 + i];
    for (int i = 0; i < 16; ++i) Pf[i]  = pf[(size_t)t * 16 + i];
    float mup[4];
    for (int i = 0; i < 4; ++i) {
      float s = b[i];
      for (int j = 0; j < 4; ++j) s += A[i*4+j] * muf[j];
      mup[i] = s;
    }
    float AP[16], Pp[16];
    mm4(A, Pf, AP);
    for (int i = 0; i < 4; ++i)
      for (int j = 0; j < 4; ++j) {
        float s = Q[i*4+j];
        for (int k = 0; k < 4; ++k) s += AP[i*4+k] * A[j*4+k];
        Pp[i*4+j] = s;
      }
    float Ppinv[16];
    spd_inv4(Pp, Ppinv, nullptr);
    float PfAT[16];
    for (int i = 0; i < 4; ++i)
      for (int j = 0; j < 4; ++j) {
        float s = 0.f;
        for (int k = 0; k < 4; ++k) s += Pf[i*4+k] * A[j*4+k];
        PfAT[i*4+j] = s;
      }
    float Gn[16];
    mm4(PfAT, Ppinv, Gn);
    float dm[4];
    for (int i = 0; i < 4; ++i) dm[i] = mus[i] - mup[i];
    float mun[4];
    for (int i = 0; i < 4; ++i) {
      float s = muf[i];
      for (int j = 0; j < 4; ++j) s += Gn[i*4+j] * dm[j];
      mun[i] = s;
    }
    float dP[16];
    for (int i = 0; i < 16; ++i) dP[i] = Ps[i] - Pp[i];
    float GdP[16];
    mm4(Gn, dP, GdP);
    float Pn[16];
    for (int i = 0; i < 4; ++i)
      for (int j = 0; j < 4; ++j) {
        float s = Pf[i*4+j];
        for (int k = 0; k < 4; ++k) s += GdP[i*4+k] * Gn[j*4+k];
        Pn[i*4+j] = s;
      }
    for (int i = 0; i < 4; ++i)  { mq[(size_t)t * 4 + i]  = mun[i]; mus[i] = mun[i]; }
    for (int i = 0; i < 16; ++i) { pq[(size_t)t * 16 + i] = Pn[i];  Ps[i]  = Pn[i];  }
  }
}

// ---------------- kernel: update beta (one block per t, lane = k) ----------------
__global__ __launch_bounds__(128) void svae_beta_kernel(
    const float* __restrict__ xhat, const float* __restrict__ mq,
    const float* __restrict__ pq, const float* __restrict__ cst,
    float* __restrict__ beta) {
  int t = blockIdx.x;
  int tid = threadIdx.x;
  __shared__ float s_part[4];
  float Qinv[16], Cinv[16];
  for (int i = 0; i < 16; ++i) { Qinv[i] = cst[CST_QINV + i]; Cinv[i] = cst[CST_CINV + i]; }
  float clp = cst[CST_CLP];
  float tr = 0.f;
  for (int i = 0; i < 4; ++i)
    for (int j = 0; j < 4; ++j) tr += Qinv[i*4+j] * pq[(size_t)t * 16 + j*4 + i];
  float mqv[4];
  for (int i = 0; i < 4; ++i) mqv[i] = mq[(size_t)t * 4 + i];
  const float* yk = xhat + (size_t)(t + 1) * KOBS * DD + tid * DD;
  float d[4];
  for (int j = 0; j < 4; ++j) d[j] = yk[j] - mqv[j];
  float quad = 0.f;
  for (int i = 0; i < 4; ++i) {
    float s = 0.f;
    for (int j = 0; j < 4; ++j) s += Cinv[i*4+j] * d[j];
    quad += d[i] * s;
  }
  float lp = -0.5f * quad + clp;
  float un = (0.9f / 128.0f) * expf(lp - 0.5f * tr);
  float s = wave_sum(un);
  if ((tid & 31) == 0) s_part[tid >> 5] = s;
  __syncthreads();
  float Z = 0.001f + s_part[0] + s_part[1] + s_part[2] + s_part[3];
  float invZ = 1.f / Z;
  if (tid == 0) beta[(size_t)t * (KOBS + 1)] = 0.001f * invZ;
  beta[(size_t)t * (KOBS + 1) + 1 + tid] = un * invZ;
}

// ---------------- kernel: sample z_recon + decoder GEMM via WMMA 16x16x4 ----------------
__global__ __launch_bounds__(256) void svae_recon_wmma_kernel(
    const float* __restrict__ mq, const float* __restrict__ pq,
    const float* __restrict__ decW, const float* __restrict__ decb,
    float* __restrict__ xrec, float* __restrict__ zrec) {
  int tid = threadIdx.x;
  int wv = tid >> 5, lane = tid & 31, n = lane & 15;
  int tile = blockIdx.x * 8 + wv;     // 128 tiles cover 2047 rows
  float z0 = 0.f, z1 = 0.f, z2 = 0.f, z3 = 0.f;
  int row = tile * 16 + n;
  bool ownrow = (lane < 16) && (row < TM1);
  if (ownrow) {
    float Pl[16];
    for (int i = 0; i < 16; ++i) Pl[i] = pq[(size_t)row * 16 + i];
    Pl[0] += 1e-6f; Pl[5] += 1e-6f; Pl[10] += 1e-6f; Pl[15] += 1e-6f;
    float L[16];
    chol4(Pl, L);
    unsigned w0, w1, w2, w3;
    threefry2x32(0u, 1u, 0u, (unsigned)(row * 2),     w0, w1);
    threefry2x32(0u, 1u, 0u, (unsigned)(row * 2 + 1), w2, w3);
    float e0, e1, e2, e3;
    boxmuller(w0, w1, e0, e1);
    boxmuller(w2, w3, e2, e3);
    z0 = mq[row*4+0] + L[0]  * e0;
    z1 = mq[row*4+1] + L[4]  * e0 + L[5]  * e1;
    z2 = mq[row*4+2] + L[8]  * e0 + L[9]  * e1 + L[10] * e2;
    z3 = mq[row*4+3] + L[12] * e0 + L[13] * e1 + L[14] * e2 + L[15] * e3;
    zrec[row*4+0] = z0; zrec[row*4+1] = z1; zrec[row*4+2] = z2; zrec[row*4+3] = z3;
  }
  // marshal per-lane z rows into WMMA A layout (rows live in lanes 0-15)
  int srcl = lane & 15;
  float s0 = __shfl(z0, srcl, 32);
  float s1 = __shfl(z1, srcl, 32);
  float s2 = __shfl(z2, srcl, 32);
  float s3 = __shfl(z3, srcl, 32);
  v2f a;
  a.x = (lane < 16) ? s0 : s2;   // K=0 | K=2
  a.y = (lane < 16) ? s1 : s3;   // K=1 | K=3
  // B: dec_W is (4,2) row-major -> B[k][n] = decW[k*2+n], cols>=2 zero
  v2f bm;
  if (lane < 16) {
    bm.x = (n < POSN) ? decW[0 * POSN + n] : 0.f;   // k=0
    bm.y = (n < POSN) ? decW[1 * POSN + n] : 0.f;   // k=1
  } else {
    bm.x = (n < POSN) ? decW[2 * POSN + n] : 0.f;   // k=2
    bm.y = (n < POSN) ? decW[3 * POSN + n] : 0.f;   // k=3
  }
  v8f c;
  float bias = (n < POSN) ? decb[n] : 0.f;
  for (int j = 0; j < 8; ++j) c[j] = bias;
  c = __builtin_amdgcn_wmma_f32_16x16x4_f32(false, a, false, bm, (short)0, c, false, false);
  if (n < POSN) {
    for (int j = 0; j < 8; ++j) {
      int M = j + ((lane < 16) ? 0 : 8);
      int r2 = tile * 16 + M;
      if (r2 < TM1) xrec[(size_t)r2 * POSN + n] = c[j];
    }
  }
}

// ---------------- host orchestration ----------------
extern "C" void kernel_launch(void* const* d_in, const int* in_sizes, int n_in,
                              void* d_out, int out_size, void* d_ws, size_t ws_size,
                              hipStream_t stream) {
  (void)in_sizes; (void)n_in; (void)out_size; (void)ws_size;
  const float* x    = (const float*)d_in[0];
  const float* encW = (const float*)d_in[1];
  const float* encb = (const float*)d_in[2];
  // d_in[3], d_in[4]: enc_W_logvar / enc_b_logvar -- computed but unused by outputs
  const float* decW = (const float*)d_in[5];
  const float* decb = (const float*)d_in[6];
  const float* A    = (const float*)d_in[7];
  const float* bkf  = (const float*)d_in[8];
  const float* Qp   = (const float*)d_in[9];
  const float* Rp   = (const float*)d_in[10];

  float* out = (float*)d_out;
  float* ws  = (float*)d_ws;
  float* xhat  = ws + WS_XHAT;
  float* betaw = ws + WS_BETA;
  float* mfw   = ws + WS_MF;
  float* pfw   = ws + WS_PF;
  float* mqw   = ws + WS_MQ;
  float* pqw   = ws + WS_PQ;
  float* cst   = ws + WS_CST;
  float* dum   = ws + WS_DUM;

  svae_setup_kernel<<<1, 1, 0, stream>>>(A, bkf, Qp, Rp, cst);
  svae_enc_wmma_kernel<<<2048, 256, 0, stream>>>(x, encW, encb, xhat);
  svae_init_beta_kernel<<<(TM1 * (KOBS + 1) + 255) / 256, 256, 0, stream>>>(betaw);

  for (int it = 0; it < ITERS; ++it) {
    float* llo = (it == ITERS - 1) ? (out + OUT_LL) : dum;
    svae_fwd_kernel<<<1, 128, 0, stream>>>(xhat, betaw, cst, mfw, pfw, llo);
    svae_bwd_kernel<<<1, 32, 0, stream>>>(mfw, pfw, cst, mqw, pqw);
    if (it < ITERS - 1)
      svae_beta_kernel<<<TM1, 128, 0, stream>>>(xhat, mqw, pqw, cst, betaw);
  }

  svae_recon_wmma_kernel<<<16, 256, 0, stream>>>(mqw, pqw, decW, decb,
                                                 out + OUT_XREC, out + OUT_ZREC);

  hipMemcpyAsync(out + OUT_MQ, mqw, sizeof(float) * (size_t)TM1 * 4,
                 hipMemcpyDeviceToDevice, stream);
  hipMemcpyAsync(out + OUT_PQ, pqw, sizeof(float) * (size_t)TM1 * 16,
                 hipMemcpyDeviceToDevice, stream);
  hipMemcpyAsync(out + OUT_MF, mfw, sizeof(float) * (size_t)TM1 * 4,
                 hipMemcpyDeviceToDevice, stream);
  hipMemcpyAsync(out + OUT_PF, pfw, sizeof(float) * (size_t)TM1 * 16,
                 hipMemcpyDeviceToDevice, stream);
}